// QuantumCBOW_18786186952934
// MI455X (gfx1250) — compile-verified
//
#include <hip/hip_runtime.h>
#include <hip/hip_bf16.h>

#define QDIM   8
#define QTRIL  36
#define QVOCAB 50257
#define QBATCH 16384
#define QW     10
#define QEPS   1e-10f
#define QALPHA 0.1f
#define WGSZ   64   // 2 waves of 32; 64 matrices staged per workgroup

typedef float v2f_t __attribute__((ext_vector_type(2)));
typedef float v8f_t __attribute__((ext_vector_type(8)));

// ---------------------------------------------------------------------------
// Cyclic Jacobi eigensolver for symmetric 8x8 (row-major flat 64 floats).
// Destroys A (diagonal converges to eigenvalues). If V != nullptr, accumulates
// eigenvectors as columns: A_in = V * diag(evals) * V^T.
// k-loops unrolled (constant indices -> registers where possible); pivot loops
// kept rolled to bound code size.
// ---------------------------------------------------------------------------
__device__ void jacobi8(float* A, float* V) {
  if (V) {
    #pragma unroll
    for (int i = 0; i < QDIM; ++i)
      #pragma unroll
      for (int j = 0; j < QDIM; ++j)
        V[i * QDIM + j] = (i == j) ? 1.f : 0.f;
  }
  #pragma unroll 1
  for (int sweep = 0; sweep < 6; ++sweep) {
    #pragma unroll 1
    for (int p = 0; p < QDIM - 1; ++p) {
      #pragma unroll 1
      for (int q = p + 1; q < QDIM; ++q) {
        float apq = A[p * QDIM + q];
        if (fabsf(apq) < 1e-20f) continue;
        float app = A[p * QDIM + p];
        float aqq = A[q * QDIM + q];
        float tau = (aqq - app) / (2.f * apq);
        float t   = copysignf(1.f, tau) / (fabsf(tau) + sqrtf(1.f + tau * tau));
        float c   = rsqrtf(1.f + t * t);
        float s   = t * c;
        #pragma unroll
        for (int k = 0; k < QDIM; ++k) {
          float akp = A[k * QDIM + p], akq = A[k * QDIM + q];
          A[k * QDIM + p] = c * akp - s * akq;
          A[k * QDIM + q] = s * akp + c * akq;
        }
        #pragma unroll
        for (int k = 0; k < QDIM; ++k) {
          float apk = A[p * QDIM + k], aqk = A[q * QDIM + k];
          A[p * QDIM + k] = c * apk - s * aqk;
          A[q * QDIM + k] = s * apk + c * aqk;
        }
        if (V) {
          #pragma unroll
          for (int k = 0; k < QDIM; ++k) {
            float vkp = V[k * QDIM + p], vkq = V[k * QDIM + q];
            V[k * QDIM + p] = c * vkp - s * vkq;
            V[k * QDIM + q] = s * vkp + c * vkq;
          }
        }
      }
    }
  }
}

// ---------------------------------------------------------------------------
// Batched 8x8 x 8x8 matmul on the matrix core: C[e] = A[e] * B[e] for all 64
// matrices staged in LDS (flat [elem][8][8], row-major). Two independent 8x8
// products are packed block-diagonally into one 16x16 WMMA tile:
//   bigA rows 0-7 = A[e0], rows 8-15 = A[e1] (K = 0..7)
//   bigB cols 0-7 = B[e0], cols 8-15 = B[e1]
// -> diagonal 8x8 blocks of D are the two wanted products; K=8 is achieved by
// chaining two V_WMMA_F32_16X16X4_F32 through the C operand.
// Uniform control flow around the WMMA ops (EXEC must be all ones).
// ---------------------------------------------------------------------------
__device__ __forceinline__ void wmma_pair_pass(const float* __restrict__ Ab,
                                               const float* __restrict__ Bb,
                                               float* __restrict__ Cb) {
  const int lane = threadIdx.x & 31;
  const int wave = threadIdx.x >> 5;
  const int hf   = lane >> 4;   // lane half: selects K group for A/B, row+8 for D
  const int l16  = lane & 15;   // M (A) / N (B,D) index within tile
  const int rA   = l16 & 7;     // row within the 8x8 block

  #pragma unroll 1
  for (int pr = 0; pr < 16; ++pr) {
    const int e0 = wave * 32 + 2 * pr;
    const int e1 = e0 + 1;
    const int eA = (l16 < 8) ? e0 : e1;  // which 8x8 block this lane feeds

    v8f_t acc = {0.f, 0.f, 0.f, 0.f, 0.f, 0.f, 0.f, 0.f};
    #pragma unroll
    for (int ks = 0; ks < 2; ++ks) {
      const int k0 = ks * 4 + hf * 2;    // A: lane-half+vgpr select K; 2 steps cover K=0..7
      v2f_t a, b;
      a.x = Ab[eA * 64 + rA * 8 + k0];
      a.y = Ab[eA * 64 + rA * 8 + k0 + 1];
      b.x = Bb[eA * 64 + k0 * 8 + rA];
      b.y = Bb[eA * 64 + (k0 + 1) * 8 + rA];
      acc = __builtin_amdgcn_wmma_f32_16x16x4_f32(
          /*neg_a=*/false, a, /*neg_b=*/false, b,
          /*c_mod=*/(short)0, acc, /*reuse_a=*/false, /*reuse_b=*/false);
    }
    // D layout: vgpr r holds row (r + 8*hf), col l16. Keep diagonal blocks only.
    if (hf == 0 && l16 < 8) {
      #pragma unroll
      for (int r = 0; r < 8; ++r) Cb[e0 * 64 + r * 8 + l16] = acc[r];
    }
    if (hf == 1 && l16 >= 8) {
      #pragma unroll
      for (int r = 0; r < 8; ++r) Cb[e1 * 64 + r * 8 + (l16 - 8)] = acc[r];
    }
  }
}

// ---------------------------------------------------------------------------
// Phase A: one density matrix per VOCAB entry (deduped: 50257 solves instead of
// 180224). rho = sym( (L L^T + eps I + corr I) / tr ), L from tril(36) with
// clamped diagonal; corr from the min eigenvalue (Jacobi, values only).
// L L^T runs on the matrix core. Result table (12.9 MB) lives in d_ws / L2.
// ---------------------------------------------------------------------------
__global__ void __launch_bounds__(WGSZ)
qcbow_vocab_density(const float* __restrict__ table, float* __restrict__ rhoT) {
  __shared__ float sA[WGSZ * 64];
  __shared__ float sB[WGSZ * 64];
  __shared__ float sC[WGSZ * 64];
  const int tid = threadIdx.x;
  const int gid = blockIdx.x * WGSZ + tid;
  const int vid = (gid < QVOCAB) ? gid : (QVOCAB - 1);  // clamp; extras skip store
  const float* p = table + (size_t)vid * QTRIL;

  // Build L (lower-tri, diag clamped) -> stage L and L^T for the WMMA pass.
  float L[64];
  #pragma unroll
  for (int i = 0; i < 64; ++i) L[i] = 0.f;
  int k = 0;
  #pragma unroll
  for (int r = 0; r < QDIM; ++r)
    #pragma unroll
    for (int c = 0; c <= r; ++c)
      L[r * 8 + c] = p[k++];
  #pragma unroll
  for (int i = 0; i < QDIM; ++i) L[i * 8 + i] = fmaxf(L[i * 8 + i], 1e-4f);
  #pragma unroll
  for (int r = 0; r < QDIM; ++r)
    #pragma unroll
    for (int c = 0; c < QDIM; ++c) {
      sA[tid * 64 + r * 8 + c] = L[r * 8 + c];
      sB[tid * 64 + r * 8 + c] = L[c * 8 + r];
    }
  __syncthreads();
  wmma_pair_pass(sA, sB, sC);          // sC = L * L^T
  __syncthreads();

  float rho[64], work[64];
  float tr = 0.f;
  #pragma unroll
  for (int r = 0; r < QDIM; ++r)
    #pragma unroll
    for (int c = 0; c < QDIM; ++c) {
      float v = 0.5f * (sC[tid * 64 + r * 8 + c] + sC[tid * 64 + c * 8 + r]);
      if (r == c) { v += QEPS; tr += v; }
      rho[r * 8 + c] = v;
    }
  #pragma unroll
  for (int i = 0; i < 64; ++i) work[i] = rho[i];
  jacobi8(work, (float*)nullptr);
  float evmin = work[0];
  #pragma unroll
  for (int i = 1; i < QDIM; ++i) evmin = fminf(evmin, work[i * 8 + i]);
  const float corr = fmaxf(1e-6f - evmin, 0.f);
  const float inv  = 1.f / (tr + 8.f * corr + QEPS);

  if (gid < QVOCAB) {
    float* dst = rhoT + (size_t)vid * 64;
    #pragma unroll
    for (int r = 0; r < QDIM; ++r)
      #pragma unroll
      for (int c = 0; c < QDIM; ++c)
        dst[r * 8 + c] = (rho[r * 8 + c] + ((r == c) ? corr : 0.f)) * inv;
  }
}

__global__ void qcbow_zero_out(float* out) { out[0] = 0.f; }

// ---------------------------------------------------------------------------
// Phase B: per batch element — masked mean of context densities (L2-resident
// gathers), full Jacobi eigendecomp of the mean, then three WMMA matmul
// passes:  sqrtR = (Q*sqrt(L)) * Q^T ;  T = sqrtR * sigma ;  P = T * sqrtR.
// Eigenvalues of P (Jacobi, values only) give the fidelity; workgroup
// reduction + one atomic per WG accumulates the mean loss.
// ---------------------------------------------------------------------------
__global__ void __launch_bounds__(WGSZ)
qcbow_fidelity(const int* __restrict__ contexts, const int* __restrict__ targets,
               const float* __restrict__ rhoT, float* __restrict__ out) {
  __shared__ float sA[WGSZ * 64];
  __shared__ float sB[WGSZ * 64];
  __shared__ float sC[WGSZ * 64];
  __shared__ float red[WGSZ];
  const int tid = threadIdx.x;
  const int b   = blockIdx.x * WGSZ + tid;   // grid covers QBATCH exactly

  // Masked mean of context density matrices.
  float rho[64];
  #pragma unroll
  for (int i = 0; i < 64; ++i) rho[i] = 0.f;
  float cnt = 0.f;
  #pragma unroll 1
  for (int w = 0; w < QW; ++w) {
    const int tok = contexts[b * QW + w];
    if (tok != 0) {
      const float* r = rhoT + (size_t)tok * 64;
      #pragma unroll
      for (int i = 0; i < 64; ++i) rho[i] += r[i];
      cnt += 1.f;
    }
  }
  const float minv = 1.f / (cnt + QEPS);
  #pragma unroll
  for (int i = 0; i < 64; ++i) rho[i] *= minv;

  // Eigendecomposition of the mean density; sqrt via Q*diag(sqrt(|l|+eps))*Q^T.
  float work[64], V[64];
  #pragma unroll
  for (int i = 0; i < 64; ++i) work[i] = rho[i];
  jacobi8(work, V);
  float sv[QDIM];
  #pragma unroll
  for (int j = 0; j < QDIM; ++j) sv[j] = sqrtf(fabsf(work[j * 8 + j]) + QEPS);

  #pragma unroll
  for (int i = 0; i < QDIM; ++i)
    #pragma unroll
    for (int j = 0; j < QDIM; ++j) {
      sA[tid * 64 + i * 8 + j] = V[i * 8 + j] * sv[j];  // Q * diag(s)
      sB[tid * 64 + i * 8 + j] = V[j * 8 + i];          // Q^T
    }
  __syncthreads();
  wmma_pair_pass(sA, sB, sC);        // sC = sqrt(rho)
  __syncthreads();

  {  // target density -> sA
    const int tok = targets[b];
    const float* g = rhoT + (size_t)tok * 64;
    #pragma unroll
    for (int i = 0; i < 64; ++i) sA[tid * 64 + i] = g[i];
  }
  __syncthreads();
  wmma_pair_pass(sC, sA, sB);        // sB = sqrt(rho) * sigma
  __syncthreads();
  wmma_pair_pass(sB, sC, sA);        // sA = sqrt(rho) * sigma * sqrt(rho)
  __syncthreads();

  // Eigenvalues of the (symmetrized) product -> fidelity -> loss.
  float P[64];
  #pragma unroll
  for (int i = 0; i < QDIM; ++i)
    #pragma unroll
    for (int j = 0; j < QDIM; ++j)
      P[i * 8 + j] = 0.5f * (sA[tid * 64 + i * 8 + j] + sA[tid * 64 + j * 8 + i]);
  jacobi8(P, (float*)nullptr);
  float fs = 0.f;
  #pragma unroll
  for (int i = 0; i < QDIM; ++i) {
    float lam = fabsf(P[i * 8 + i]);
    lam = fminf(fmaxf(lam, QEPS), 1.f - QEPS);
    fs += sqrtf(lam);
  }
  float f = fs * fs;
  f = fminf(fmaxf(f, 0.f), 1.f);
  const float loss = -logf(fmaxf(f, 1e-8f)) + QALPHA * (1.f - f);

  red[tid] = loss;
  __syncthreads();
  if (tid == 0) {
    float acc = 0.f;
    #pragma unroll 1
    for (int i = 0; i < WGSZ; ++i) acc += red[i];
    atomicAdd(out, acc * (1.f / (float)QBATCH));
  }
}

extern "C" void kernel_launch(void* const* d_in, const int* in_sizes, int n_in,
                              void* d_out, int out_size, void* d_ws, size_t ws_size,
                              hipStream_t stream) {
  (void)in_sizes; (void)n_in; (void)out_size; (void)ws_size;
  const int*   contexts = (const int*)d_in[0];    // [B, W]
  const int*   targets  = (const int*)d_in[1];    // [B]
  const float* table    = (const float*)d_in[2];  // [VOCAB, 36]
  float* out  = (float*)d_out;                    // scalar f32
  float* rhoT = (float*)d_ws;                     // VOCAB * 64 floats (12.9 MB)

  const int gridA = (QVOCAB + WGSZ - 1) / WGSZ;
  qcbow_vocab_density<<<gridA, WGSZ, 0, stream>>>(table, rhoT);
  qcbow_zero_out<<<1, 1, 0, stream>>>(out);
  qcbow_fidelity<<<QBATCH / WGSZ, WGSZ, 0, stream>>>(contexts, targets, rhoT, out);
}